// BQWarp_49435073577128
// MI455X (gfx1250) — compile-verified
//
#include <hip/hip_runtime.h>
#include <hip/hip_bf16.h>

// ---------------------------------------------------------------------------
// Ball query (radius 0.25, K=10) fused with V_WMMA_F32_16X16X4_F32 on gfx1250.
//   points:  x      (1, 16384, 3)  f32
//   queries: p_grid (1, 32,32,16,3) -> (16384, 3) f32
//   out:     mapping (1,Q,10) int32  ++  outputs (1,Q,10,3) f32   (concat flat)
// ---------------------------------------------------------------------------

#define NPTS   16384
#define NQRY   16384
#define KNN    10
#define RADIUS2 0.0625f      // 0.25^2
#define PT_TILES (NPTS / 16)

typedef float v2f __attribute__((ext_vector_type(2)));
typedef float v8f __attribute__((ext_vector_type(8)));

// ---- pack points: pts4[i] = {x,y,z,|p|^2}, pn[i] = |p|^2 --------------------
__global__ __launch_bounds__(256)
void bq_pack_points(const float* __restrict__ x,
                    float4* __restrict__ pts4,
                    float*  __restrict__ pn)
{
    int i = blockIdx.x * blockDim.x + threadIdx.x;
    if (i < NPTS) {
        float px = x[3 * i + 0];
        float py = x[3 * i + 1];
        float pz = x[3 * i + 2];
        float s  = px * px + py * py + pz * pz;
        pts4[i] = make_float4(px, py, pz, s);
        pn[i]   = s;
    }
}

// ---- fused distance (WMMA) + top-K selection --------------------------------
// One wave (32 lanes) owns 16 queries. Loop over 1024 point tiles of 16.
//   A (16 pts x 4 coords): lanes 0-15 hold {x,y} (K=0,1), lanes 16-31 {z,0} (K=2,3)
//   B (4 x 16 queries):    lanes 0-15 hold {x,y},         lanes 16-31 {z,0}
//   D (16x16 f32): lane L<16 -> rows M=0..7 col N=L; lane L>=16 -> rows 8..15.
// Each lane keeps a sorted top-10 of its 8 rows; lane pairs (L, L^16) merge.
__global__ __launch_bounds__(256)
void bq_wmma_topk(const float* __restrict__ qry,      // (NQRY,3)
                  const float4* __restrict__ pts4,    // packed points
                  const float* __restrict__ pn,       // |p|^2
                  int*   __restrict__ map_out,        // (NQRY,KNN)
                  float* __restrict__ pts_out)        // (NQRY,KNN,3)
{
    const int lane  = threadIdx.x & 31;
    const int wave  = threadIdx.x >> 5;
    const int qtile = blockIdx.x * (blockDim.x >> 5) + wave;
    const int qcol  = lane & 15;
    const bool hi   = lane >= 16;
    const int qidx  = qtile * 16 + qcol;

    // Query for this lane's column (both half-waves load the same query).
    const float qx = qry[3 * qidx + 0];
    const float qy = qry[3 * qidx + 1];
    const float qz = qry[3 * qidx + 2];
    const float qn = qx * qx + qy * qy + qz * qz;

    v2f B;
    B[0] = hi ? qz : qx;     // K=2 : K=0
    B[1] = hi ? 0.f : qy;    // pad : K=1

    float best_d[KNN];
    int   best_i[KNN];
#pragma unroll
    for (int j = 0; j < KNN; ++j) { best_d[j] = __builtin_inff(); best_i[j] = 0; }

    const int rowbase = hi ? 8 : 0;

    for (int t = 0; t < PT_TILES; ++t) {
        const int pbase = t * 16;

        // prefetch next tile of packed points (speculative, cache-resident)
        __builtin_prefetch(&pts4[pbase + 16 + qcol], 0, 3);

        const float4 p4 = pts4[pbase + qcol];
        v2f A;
        A[0] = hi ? p4.z : p4.x;
        A[1] = hi ? 0.f  : p4.y;

        // |p|^2 for my 8 rows: two b128 loads
        const float4* pn4 = reinterpret_cast<const float4*>(pn + pbase + rowbase);
        const float4 na = pn4[0];
        const float4 nb = pn4[1];
        float qpn[8];
        qpn[0] = qn + na.x; qpn[1] = qn + na.y; qpn[2] = qn + na.z; qpn[3] = qn + na.w;
        qpn[4] = qn + nb.x; qpn[5] = qn + nb.y; qpn[6] = qn + nb.z; qpn[7] = qn + nb.w;

        v8f c = {};
        c = __builtin_amdgcn_wmma_f32_16x16x4_f32(
                /*neg_a=*/false, A, /*neg_b=*/false, B,
                /*c_mod=*/(short)0, c, /*reuse_a=*/false, /*reuse_b=*/false);

#pragma unroll
        for (int r = 0; r < 8; ++r) {
            const float d2 = fmaf(-2.0f, c[r], qpn[r]);
            if (d2 <= RADIUS2 && d2 < best_d[KNN - 1]) {
                float cd = d2;
                int   ci = pbase + rowbase + r;
#pragma unroll
                for (int j = 0; j < KNN; ++j) {          // carry-insert (branchless)
                    const bool take = cd < best_d[j];
                    const float td = best_d[j]; const int ti = best_i[j];
                    best_d[j] = take ? cd : td;
                    best_i[j] = take ? ci : ti;
                    cd = take ? td : cd;
                    ci = take ? ti : ci;
                }
            }
        }
    }

    // ---- merge lane L with partner lane L^16 (same query, disjoint rows) ----
    float pd[KNN]; int pi[KNN];
#pragma unroll
    for (int j = 0; j < KNN; ++j) {
        pd[j] = __shfl_xor(best_d[j], 16, 32);
        pi[j] = __shfl_xor(best_i[j], 16, 32);
    }
#pragma unroll
    for (int j = 0; j < KNN; ++j) {
        float cd = pd[j];
        int   ci = pi[j];
#pragma unroll
        for (int m = 0; m < KNN; ++m) {
            const bool take = cd < best_d[m];
            const float td = best_d[m]; const int ti = best_i[m];
            best_d[m] = take ? cd : td;
            best_i[m] = take ? ci : ti;
            cd = take ? td : cd;
            ci = take ? ti : ci;
        }
    }

    // ---- lanes 0..15 write results ----
    if (!hi) {
#pragma unroll
        for (int j = 0; j < KNN; ++j) {
            const bool valid = best_d[j] <= RADIUS2;     // inf -> no candidate
            const int idx = valid ? best_i[j] : 0;
            map_out[qidx * KNN + j] = idx;
            const float4 p = pts4[idx];
            const long o = (long)(qidx * KNN + j) * 3;
            pts_out[o + 0] = valid ? p.x : 0.f;
            pts_out[o + 1] = valid ? p.y : 0.f;
            pts_out[o + 2] = valid ? p.z : 0.f;
        }
    }
}

// ---------------------------------------------------------------------------
extern "C" void kernel_launch(void* const* d_in, const int* in_sizes, int n_in,
                              void* d_out, int out_size, void* d_ws, size_t ws_size,
                              hipStream_t stream)
{
    (void)in_sizes; (void)n_in; (void)out_size; (void)ws_size;

    const float* x      = (const float*)d_in[0];   // (1,16384,3)
    const float* p_grid = (const float*)d_in[1];   // (1,32,32,16,3) -> (Q,3)

    // workspace layout: [0, 256KB) packed float4 points ; [256KB, 320KB) |p|^2
    float4* pts4 = (float4*)d_ws;
    float*  pn   = (float*)((char*)d_ws + (size_t)NPTS * sizeof(float4));

    int*   map_out = (int*)d_out;                         // Q*KNN int32 (bit-compatible slot)
    float* pts_out = (float*)d_out + (size_t)NQRY * KNN;  // Q*KNN*3 f32

    bq_pack_points<<<(NPTS + 255) / 256, 256, 0, stream>>>(x, pts4, pn);

    // 8 waves / block, each wave owns a 16-query tile: 1024 tiles -> 128 blocks
    bq_wmma_topk<<<NQRY / 16 / 8, 256, 0, stream>>>(p_grid, pts4, pn, map_out, pts_out);
}